// ExponentialSmoother_30580167147914
// MI455X (gfx1250) — compile-verified
//
#include <hip/hip_runtime.h>
#include <math.h>

// ExponentialSmoother: out[b,n] = sum_t S[b,t,n] * w[t],  w[t] = e^{-t/20} / Z
// S: [B=64, T=1000, N=4096] f32 row-major (n fastest), out: [B, N] f32.
//
// HBM-bound (1.05 GB @ 23.3 TB/s ~ 45 us). Per-batch GEMV mapped onto
// V_WMMA_F32_16X16X4_F32 with the weight row replicated across A's 16 rows:
//   A[m,k] = w[t0+k] (all m identical) -> every row of C = desired 16-wide slice.
// 16x redundant f32 WMMA FLOPs (~186 TF issued at full BW) stays far below the
// matrix ceiling, so the kernel remains memory-bound with exact f32 numerics.
//
// N is a compile-time constant so the 4 per-chunk loads share one address
// register with immediate IOFFSETs (0 / 16 KB / 64 KB / 80 KB), and the loop
// advances the pointer with a single 64-bit add of 128 KB.

typedef __attribute__((ext_vector_type(2))) float v2f;
typedef __attribute__((ext_vector_type(8))) float v8f;

#define WAVES_PER_BLOCK 8

__global__ __launch_bounds__(256) void
ExponentialSmoother_wmma(const float* __restrict__ S, float* __restrict__ out,
                         int T, int totalTiles, float d1 /* e^{-1/20} */, float invZ) {
    constexpr int N = 4096;                  // neuron dim (reference shape)
    constexpr int TILES_PER_BATCH = N / 16;  // 256 -> shifts/masks, no div

    const int tid  = threadIdx.x;
    const int wave = tid >> 5;
    const int lane = tid & 31;

    const int tile = blockIdx.x * WAVES_PER_BLOCK + wave;
    if (tile >= totalTiles) return;          // whole-wave exit: EXEC stays all-1s for WMMA

    const int b    = tile / TILES_PER_BATCH;        // tile >> 8
    const int n0   = (tile & (TILES_PER_BATCH - 1)) << 4;
    const int half = lane >> 4;              // 0: K={0,1}, 1: K={2,3} (A 16x4 f32 layout)
    const int col  = lane & 15;              // N index within tile (B/C/D lane = column)

    // Decay constants
    const float d2 = d1 * d1;
    const float d4 = d2 * d2;
    const float d8 = d4 * d4;

    // A fragment: A[m,k] = w[t0+k] * (1/Z), identical for all 16 rows m.
    // lanes 0-15: {K0,K1} in {a.x,a.y}; lanes 16-31: {K2,K3}.
    v2f a0, a1;
    a0.x = invZ * (half ? d2 : 1.0f);
    a0.y = a0.x * d1;
    a1.x = a0.x * d4;                        // second accumulator: t0+4 chunk
    a1.y = a0.y * d4;

    // B fragment source: lane (half,col) reads rows t0+2*half, t0+2*half+1 at column n0+col.
    const float* p = S + (size_t)b * (size_t)T * N + (size_t)(half * 2) * N + n0 + col;

    v8f c0 = {};
    v8f c1 = {};

    // Software pipeline: next chunk-pair's loads in flight during the WMMAs.
    v2f b0, b1;
    b0.x = p[0];     b0.y = p[N];            // IOFFSET 0, 16384 B
    b1.x = p[4 * N]; b1.y = p[5 * N];        // IOFFSET 65536, 81920 B
    p += 8 * N;

    const int iters = T / 8;                 // 125 for T=1000
    for (int i = 0; i < iters - 1; ++i) {
        v2f nb0, nb1;
        nb0.x = p[0];     nb0.y = p[N];
        nb1.x = p[4 * N]; nb1.y = p[5 * N];
        p += 8 * N;

        // D = A*B + C   (8 args: neg_a, A, neg_b, B, c_mod, C, reuse_a, reuse_b)
        c0 = __builtin_amdgcn_wmma_f32_16x16x4_f32(false, a0, false, b0, (short)0, c0, false, false);
        c1 = __builtin_amdgcn_wmma_f32_16x16x4_f32(false, a1, false, b1, (short)0, c1, false, false);

        // advance weights by 8 time steps (v_pk_mul_f32)
        a0.x *= d8; a0.y *= d8;
        a1.x *= d8; a1.y *= d8;
        b0 = nb0; b1 = nb1;
    }
    c0 = __builtin_amdgcn_wmma_f32_16x16x4_f32(false, a0, false, b0, (short)0, c0, false, false);
    c1 = __builtin_amdgcn_wmma_f32_16x16x4_f32(false, a1, false, b1, (short)0, c1, false, false);

    // C/D layout: VGPR0, lanes 0-15 = row M=0 (all rows identical by construction).
    if (lane < 16) {
        out[(size_t)b * N + n0 + lane] = c0[0] + c1[0];
    }
}

extern "C" void kernel_launch(void* const* d_in, const int* in_sizes, int n_in,
                              void* d_out, int out_size, void* d_ws, size_t ws_size,
                              hipStream_t stream) {
    const float* S = (const float*)d_in[0];
    float* out = (float*)d_out;

    const int N = 4096;                       // must match kernel constexpr
    const int B = out_size / N;               // 64
    const int T = in_sizes[0] / out_size;     // 1000

    const int totalTiles = B * (N / 16);      // 16384 waves

    // Normalization 1/Z with Z = sum_{t<T} r^t = (1 - r^T) / (1 - r), r = e^{-1/20}
    double r = exp(-1.0 / 20.0);
    double Z = (1.0 - pow(r, (double)T)) / (1.0 - r);
    float invZ = (float)(1.0 / Z);
    float d1 = (float)r;

    int blocks = (totalTiles + WAVES_PER_BLOCK - 1) / WAVES_PER_BLOCK;
    ExponentialSmoother_wmma<<<blocks, 256, 0, stream>>>(S, out, T, totalTiles, d1, invZ);
}